// GatingGCN_34703335751945
// MI455X (gfx1250) — compile-verified
//
#include <hip/hip_runtime.h>
#include <hip/hip_bf16.h>
#include <math.h>

typedef __attribute__((ext_vector_type(2))) float v2f;
typedef __attribute__((ext_vector_type(8))) float v8f;

#define H 128
#define NGRAPH 64
#define NEXP 8

// ---------------------------------------------------------------------------
// CSR build (once per launch): dst-major adjacency with self-loop in slot 0.
// ---------------------------------------------------------------------------
__global__ void k_count(const int* __restrict__ dst, int* __restrict__ cnt, int E) {
    int e = blockIdx.x * 256 + threadIdx.x;
    if (e < E) atomicAdd(&cnt[dst[e]], 1);
}

// Single-block scan over N (exclusive prefix of cnt[i]+1) + deg^-1/2.
__global__ void k_scan(const int* __restrict__ cnt, int* __restrict__ rowptr,
                       float* __restrict__ dis, int N) {
    __shared__ int sh[1024];
    __shared__ int base_sum;
    int t = threadIdx.x;
    if (t == 0) base_sum = 0;
    __syncthreads();
    for (int base = 0; base < N; base += 1024) {
        int i = base + t;
        int v = (i < N) ? (cnt[i] + 1) : 0;   // +1 = self loop
        sh[t] = v;
        __syncthreads();
        for (int offs = 1; offs < 1024; offs <<= 1) {
            int x = (t >= offs) ? sh[t - offs] : 0;
            __syncthreads();
            sh[t] += x;
            __syncthreads();
        }
        if (i < N) {
            rowptr[i] = base_sum + sh[t] - v;      // exclusive prefix
            dis[i]    = rsqrtf((float)v);          // deg >= 1 always
        }
        int total = sh[1023];
        __syncthreads();
        if (t == 0) base_sum += total;
        __syncthreads();
    }
    if (t == 0) rowptr[N] = base_sum;
}

__global__ void k_place_self(const int* __restrict__ rowptr, const float* __restrict__ dis,
                             int* __restrict__ csr_s, float* __restrict__ csr_w, int N) {
    int n = blockIdx.x * 256 + threadIdx.x;
    if (n < N) {
        int p = rowptr[n];
        float r = dis[n];
        csr_s[p] = n;
        csr_w[p] = r * r;
    }
}

__global__ void k_place_edges(const int* __restrict__ src, const int* __restrict__ dst,
                              const int* __restrict__ rowptr, int* __restrict__ cnt,
                              const float* __restrict__ dis,
                              int* __restrict__ csr_s, float* __restrict__ csr_w, int E) {
    int e = blockIdx.x * 256 + threadIdx.x;
    if (e < E) {
        int s = src[e], d = dst[e];
        int p = rowptr[d] + 1 + atomicAdd(&cnt[d], 1);   // slot 0 = self loop
        csr_s[p] = s;
        csr_w[p] = dis[s] * dis[d];
    }
}

// Shared C-fragment store: uniform (scalar-branch) tile-bounds check so the
// fast path keeps EXEC all-ones and the 8 stores can clause.
__device__ __forceinline__ void store_ctile(float* __restrict__ out, const v8f& c,
                                            int tile_m, int half, int col, int N) {
    int rb = tile_m + half * 8;
    if (tile_m + 16 <= N) {            // scalar condition -> s_cbranch, no exec mask
#pragma unroll
        for (int v = 0; v < 8; v++)
            out[(size_t)(rb + v) * H + col] = c[v];
    } else {
#pragma unroll
        for (int v = 0; v < 8; v++)
            if (rb + v < N) out[(size_t)(rb + v) * H + col] = c[v];
    }
}

// ---------------------------------------------------------------------------
// GEMM layer 0: h = [Z | pos] @ W0   (A is 16x4 f32, single WMMA per tile)
// block = 256 (8 waves), wave w -> 16x16 tile at (blockIdx.x*16, 16w)
// ---------------------------------------------------------------------------
__global__ void k_gemm0(const float* __restrict__ Z, const float* __restrict__ pos,
                        const float* __restrict__ W0, float* __restrict__ out, int N) {
    int wave = threadIdx.x >> 5, lane = threadIdx.x & 31;
    int half = lane >> 4, l = lane & 15;
    int tile_m = blockIdx.x * 16;
    int row = min(tile_m + l, N - 1);
    int col = wave * 16 + l;
    int ka = half * 2;                               // K = vgpr + 2*(lane>=16)
    v2f a, b;
    a.x = (ka == 0) ? Z[row] : pos[row * 3 + (ka - 1)];
    a.y = pos[row * 3 + ka];                         // feature ka+1 -> pos[ka]
    b.x = W0[ka * H + col];
    b.y = W0[(ka + 1) * H + col];
    v8f c = {};
    c = __builtin_amdgcn_wmma_f32_16x16x4_f32(false, a, false, b, (short)0, c, false, false);
    store_ctile(out, c, tile_m, half, col, N);
}

// ---------------------------------------------------------------------------
// GEMM layers 1/2: h = relu(agg_prev + b_prev) @ W   (bias+relu fused at A-load)
// K = 128 -> 32 x V_WMMA_F32_16X16X4_F32 per tile.
// ---------------------------------------------------------------------------
__global__ void k_gemm(const float* __restrict__ in, const float* __restrict__ bias,
                       const float* __restrict__ W, float* __restrict__ out, int N) {
    int wave = threadIdx.x >> 5, lane = threadIdx.x & 31;
    int half = lane >> 4, l = lane & 15;
    int tile_m = blockIdx.x * 16;
    int row = min(tile_m + l, N - 1);
    int col = wave * 16 + l;
    const float* arow = in + (size_t)row * H;
    v8f c = {};
#pragma unroll 8
    for (int k0 = 0; k0 < H; k0 += 4) {
        int ka = k0 + half * 2;
        float2 av = *(const float2*)(arow + ka);
        float2 bv = *(const float2*)(bias + ka);
        v2f a, b;
        a.x = fmaxf(av.x + bv.x, 0.0f);
        a.y = fmaxf(av.y + bv.y, 0.0f);
        b.x = W[ka * H + col];
        b.y = W[(ka + 1) * H + col];
        c = __builtin_amdgcn_wmma_f32_16x16x4_f32(false, a, false, b, (short)0, c, false, false);
    }
    store_ctile(out, c, tile_m, half, col, N);
}

// ---------------------------------------------------------------------------
// Aggregation: one wave per node, 32 lanes x float4 = 128 columns.
// Register accumulation, single store, no float atomics. h stays L2-resident.
// ---------------------------------------------------------------------------
__global__ void k_aggregate(const float* __restrict__ h, const int* __restrict__ rowptr,
                            const int* __restrict__ csr_s, const float* __restrict__ csr_w,
                            float* __restrict__ out, int N) {
    int node = blockIdx.x * 8 + (threadIdx.x >> 5);
    int lane = threadIdx.x & 31;
    if (node >= N) return;
    int beg = rowptr[node], end = rowptr[node + 1];
    float4 acc = {0.f, 0.f, 0.f, 0.f};
    for (int j = beg; j < end; j++) {
        int s = csr_s[j];
        float w = csr_w[j];
        if (j + 1 < end) {                      // prefetch next neighbor row
            int sn = csr_s[j + 1];
            __builtin_prefetch(h + (size_t)sn * H + lane * 4, 0, 0);
        }
        float4 v = *(const float4*)(h + (size_t)s * H + lane * 4);
        acc.x += w * v.x; acc.y += w * v.y; acc.z += w * v.z; acc.w += w * v.w;
    }
    *(float4*)(out + (size_t)node * H + lane * 4) = acc;
}

// ---------------------------------------------------------------------------
// Global mean pool: relu(agg+b2), atomic add into 64x128 slots + counts.
// ---------------------------------------------------------------------------
__global__ void k_pool(const float* __restrict__ x, const float* __restrict__ bias,
                       const int* __restrict__ batch, float* __restrict__ pooled,
                       float* __restrict__ gcnt, int N) {
    int node = blockIdx.x * 8 + (threadIdx.x >> 5);
    int lane = threadIdx.x & 31;
    if (node >= N) return;
    int g = batch[node];
    int cidx = lane * 4;
    float4 v  = *(const float4*)(x + (size_t)node * H + cidx);
    float4 b4 = *(const float4*)(bias + cidx);
    float* p = pooled + g * H + cidx;
    atomicAdd(p + 0, fmaxf(v.x + b4.x, 0.0f));
    atomicAdd(p + 1, fmaxf(v.y + b4.y, 0.0f));
    atomicAdd(p + 2, fmaxf(v.z + b4.z, 0.0f));
    atomicAdd(p + 3, fmaxf(v.w + b4.w, 0.0f));
    if (lane == 0) atomicAdd(&gcnt[g], 1.0f);
}

// ---------------------------------------------------------------------------
// Head: pooled/count @ lin_W + lin_b, softmax over 8 experts. 64 threads.
// ---------------------------------------------------------------------------
__global__ void k_head(const float* __restrict__ pooled, const float* __restrict__ gcnt,
                       const float* __restrict__ linW, const float* __restrict__ linb,
                       float* __restrict__ out) {
    int g = threadIdx.x;
    if (g >= NGRAPH) return;
    float inv = 1.0f / fmaxf(gcnt[g], 1.0f);
    float logits[NEXP];
#pragma unroll
    for (int j = 0; j < NEXP; j++) logits[j] = linb[j];
    for (int k = 0; k < H; k++) {
        float xv = pooled[g * H + k] * inv;
#pragma unroll
        for (int j = 0; j < NEXP; j++) logits[j] += xv * linW[k * NEXP + j];
    }
    float mx = logits[0];
#pragma unroll
    for (int j = 1; j < NEXP; j++) mx = fmaxf(mx, logits[j]);
    float s = 0.0f, e[NEXP];
#pragma unroll
    for (int j = 0; j < NEXP; j++) { e[j] = __expf(logits[j] - mx); s += e[j]; }
    float si = 1.0f / s;
#pragma unroll
    for (int j = 0; j < NEXP; j++) out[g * NEXP + j] = e[j] * si;
}

// ---------------------------------------------------------------------------
extern "C" void kernel_launch(void* const* d_in, const int* in_sizes, int n_in,
                              void* d_out, int out_size, void* d_ws, size_t ws_size,
                              hipStream_t stream) {
    const float* Z    = (const float*)d_in[0];
    const float* pos  = (const float*)d_in[1];
    const float* W0   = (const float*)d_in[2];
    const float* b0   = (const float*)d_in[3];
    const float* W1   = (const float*)d_in[4];
    const float* b1   = (const float*)d_in[5];
    const float* W2   = (const float*)d_in[6];
    const float* b2   = (const float*)d_in[7];
    const float* linW = (const float*)d_in[8];
    const float* linb = (const float*)d_in[9];
    const int*   eidx = (const int*)d_in[10];
    const int*   batch= (const int*)d_in[11];
    const int N = in_sizes[0];
    const int E = in_sizes[10] / 2;
    const int* src = eidx;
    const int* dst = eidx + E;

    char* ws = (char*)d_ws;
    size_t off = 0;
    auto take = [&](size_t bytes) -> void* {
        void* p = ws + off;
        off += (bytes + 255) & ~(size_t)255;
        return p;
    };
    int*   cnt    = (int*)  take((size_t)N * 4);
    int*   rowptr = (int*)  take((size_t)(N + 1) * 4);
    float* dis    = (float*)take((size_t)N * 4);
    int*   csr_s  = (int*)  take((size_t)(E + N) * 4);
    float* csr_w  = (float*)take((size_t)(E + N) * 4);
    float* bufA   = (float*)take((size_t)N * H * 4);   // GEMM output h
    float* bufB   = (float*)take((size_t)N * H * 4);   // aggregation output
    float* pooled = (float*)take((size_t)NGRAPH * H * 4);
    float* gcnt   = (float*)take((size_t)NGRAPH * 4);

    dim3 blk(256);
    int eb = (E + 255) / 256;
    int nb = (N + 255) / 256;
    int tb = (N + 15) / 16;     // one 16-row tile per block (8 waves x 16 cols)
    int wb = (N + 7) / 8;       // one wave per node

    // CSR build
    hipMemsetAsync(cnt, 0, (size_t)N * 4, stream);
    k_count<<<eb, blk, 0, stream>>>(dst, cnt, E);
    k_scan<<<1, 1024, 0, stream>>>(cnt, rowptr, dis, N);
    hipMemsetAsync(cnt, 0, (size_t)N * 4, stream);
    k_place_self<<<nb, blk, 0, stream>>>(rowptr, dis, csr_s, csr_w, N);
    k_place_edges<<<eb, blk, 0, stream>>>(src, dst, rowptr, cnt, dis, csr_s, csr_w, E);

    // Layer 0
    k_gemm0<<<tb, blk, 0, stream>>>(Z, pos, W0, bufA, N);
    k_aggregate<<<wb, blk, 0, stream>>>(bufA, rowptr, csr_s, csr_w, bufB, N);
    // Layer 1 (bias0 + relu fused into A-load)
    k_gemm<<<tb, blk, 0, stream>>>(bufB, b0, W1, bufA, N);
    k_aggregate<<<wb, blk, 0, stream>>>(bufA, rowptr, csr_s, csr_w, bufB, N);
    // Layer 2
    k_gemm<<<tb, blk, 0, stream>>>(bufB, b1, W2, bufA, N);
    k_aggregate<<<wb, blk, 0, stream>>>(bufA, rowptr, csr_s, csr_w, bufB, N);

    // Pool (bias2 + relu fused) + head
    hipMemsetAsync(pooled, 0, (size_t)NGRAPH * H * 4, stream);
    hipMemsetAsync(gcnt, 0, (size_t)NGRAPH * 4, stream);
    k_pool<<<wb, blk, 0, stream>>>(bufB, b2, batch, pooled, gcnt, N);
    k_head<<<1, 64, 0, stream>>>(pooled, gcnt, linW, linb, (float*)d_out);
}